// transformation_loss_62629213110365
// MI455X (gfx1250) — compile-verified
//
#include <hip/hip_runtime.h>
#include <hip/hip_bf16.h>

typedef __attribute__((ext_vector_type(2))) float v2f;
typedef __attribute__((ext_vector_type(8))) float v8f;

#define P_PTS 512
#define M_ANCH 128
#define NTHREADS 256
#define PI_F 3.14159265358979323846f

__device__ __forceinline__ float3 rodrigues(float3 p, float3 axis, float cos_a, float sin_a) {
    // r = p*cos + cross(axis,p)*sin + axis*(axis.p)*(1-cos)
    float one_c = 1.0f - cos_a;
    float adp = axis.x * p.x + axis.y * p.y + axis.z * p.z;
    float cx = axis.y * p.z - axis.z * p.y;
    float cy = axis.z * p.x - axis.x * p.z;
    float cz = axis.x * p.y - axis.y * p.x;
    float3 r;
    r.x = p.x * cos_a + cx * sin_a + axis.x * adp * one_c;
    r.y = p.y * cos_a + cy * sin_a + axis.y * adp * one_c;
    r.z = p.z * cos_a + cz * sin_a + axis.z * adp * one_c;
    return r;
}

__global__ void __launch_bounds__(NTHREADS)
anchor_distance_kernel(const float* __restrict__ reg_rot,
                       const float* __restrict__ reg_trans,
                       const float* __restrict__ tgt_rot,
                       const float* __restrict__ tgt_trans,
                       const float* __restrict__ model_points,
                       const int* __restrict__ is_symmetric,
                       const int* __restrict__ class_indices,
                       float* __restrict__ per_anchor) {
    __shared__ float4 sPred[P_PTS];
    __shared__ float4 sTarg[P_PTS];
    __shared__ float  sPn[P_PTS];
    __shared__ float  sTn[P_PTS];
    __shared__ unsigned int sMin[P_PTS];
    __shared__ float  sRed[NTHREADS];

    const int m   = blockIdx.x;
    const int tid = threadIdx.x;

    // ---- per-anchor pose (uniform across block) ----
    float3 vp, vt, tp3, tt3;
    vp.x = reg_rot[m * 3 + 0] * PI_F; vp.y = reg_rot[m * 3 + 1] * PI_F; vp.z = reg_rot[m * 3 + 2] * PI_F;
    vt.x = tgt_rot[m * 3 + 0] * PI_F; vt.y = tgt_rot[m * 3 + 1] * PI_F; vt.z = tgt_rot[m * 3 + 2] * PI_F;
    tp3.x = reg_trans[m * 3 + 0]; tp3.y = reg_trans[m * 3 + 1]; tp3.z = reg_trans[m * 3 + 2];
    tt3.x = tgt_trans[m * 3 + 0]; tt3.y = tgt_trans[m * 3 + 1]; tt3.z = tgt_trans[m * 3 + 2];

    float ang_p = sqrtf(vp.x * vp.x + vp.y * vp.y + vp.z * vp.z);
    float ang_t = sqrtf(vt.x * vt.x + vt.y * vt.y + vt.z * vt.z);
    float inv_p = 1.0f / ang_p, inv_t = 1.0f / ang_t;
    float3 axp = {vp.x * inv_p, vp.y * inv_p, vp.z * inv_p};
    float3 axt = {vt.x * inv_t, vt.y * inv_t, vt.z * inv_t};
    float cp = cosf(ang_p), sp = sinf(ang_p);
    float ct = cosf(ang_t), st = sinf(ang_t);

    const int cls = class_indices[m];
    const int sym = is_symmetric[m];
    const float* pts = model_points + (size_t)cls * P_PTS * 3;

    // ---- transform points into LDS; accumulate paired (ADD) distance ----
    float asym_acc = 0.0f;
    for (int i = tid; i < P_PTS; i += NTHREADS) {
        float3 p;
        p.x = pts[i * 3 + 0]; p.y = pts[i * 3 + 1]; p.z = pts[i * 3 + 2];
        float3 rp = rodrigues(p, axp, cp, sp);
        rp.x += tp3.x; rp.y += tp3.y; rp.z += tp3.z;
        float3 rt = rodrigues(p, axt, ct, st);
        rt.x += tt3.x; rt.y += tt3.y; rt.z += tt3.z;

        sPred[i] = make_float4(rp.x, rp.y, rp.z, 0.0f);
        sTarg[i] = make_float4(rt.x, rt.y, rt.z, 0.0f);
        sPn[i] = rp.x * rp.x + rp.y * rp.y + rp.z * rp.z;
        sTn[i] = rt.x * rt.x + rt.y * rt.y + rt.z * rt.z;
        sMin[i] = 0x7F7FFFFFu;  // FLT_MAX bits (non-negative float ordering == uint ordering)

        float dx = rp.x - rt.x, dy = rp.y - rt.y, dz = rp.z - rt.z;
        asym_acc += sqrtf(dx * dx + dy * dy + dz * dz);
    }
    __syncthreads();

    float result;
    if (sym == 1) {
        // ---- ADD-S: pairwise min distance via V_WMMA_F32_16X16X4_F32 ----
        // dist^2(j_targ, i_pred) = ||t_j||^2 + ||p_i||^2 - 2 * (t_j . p_i)
        // A = targ tile (16x4, xyz0), B = pred tile (4x16), D = 16x16 dots.
        // Each wave owns a row tile: A + targ norms stay in registers across
        // the full 32-column sweep (halves LDS traffic per WMMA).
        const int wave = tid >> 5;
        const int lane = tid & 31;
        const int l    = lane & 15;
        const int hi   = lane >> 4;  // 0: K=0,1 half; 1: K=2,3 half

        for (int jt = wave; jt < P_PTS / 16; jt += NTHREADS / 32) {
            const int j0 = jt << 4;
            float4 tv = sTarg[j0 + l];
            v2f a;
            a.x = hi ? tv.z : tv.x;
            a.y = hi ? 0.0f : tv.y;

            float tn[8];
            #pragma unroll
            for (int v = 0; v < 8; ++v) tn[v] = sTn[j0 + hi * 8 + v];

            #pragma unroll 4
            for (int it = 0; it < P_PTS / 16; ++it) {
                const int i0 = it << 4;
                float4 pv = sPred[i0 + l];
                v2f b;
                b.x = hi ? pv.z : pv.x;
                b.y = hi ? 0.0f : pv.y;

                v8f c = {};
                // 8 args: (neg_a, A, neg_b, B, c_mod, C, reuse_a, reuse_b)
                c = __builtin_amdgcn_wmma_f32_16x16x4_f32(false, a, false, b, (short)0, c, false, false);

                const float pn_i = sPn[i0 + l];
                float mn = 3.4e38f;
                #pragma unroll
                for (int v = 0; v < 8; ++v)
                    mn = fminf(mn, tn[v] + pn_i - 2.0f * c[v]);
                mn = fmaxf(mn, 0.0f);
                atomicMin(&sMin[i0 + l], __float_as_uint(mn));
            }
        }
        __syncthreads();

        float s = 0.0f;
        for (int i = tid; i < P_PTS; i += NTHREADS)
            s += sqrtf(__uint_as_float(sMin[i]));
        sRed[tid] = s;
        __syncthreads();
        for (int off = NTHREADS / 2; off > 0; off >>= 1) {
            if (tid < off) sRed[tid] += sRed[tid + off];
            __syncthreads();
        }
        result = sRed[0] * (1.0f / P_PTS);
    } else {
        // ---- ADD: paired mean distance ----
        sRed[tid] = asym_acc;
        __syncthreads();
        for (int off = NTHREADS / 2; off > 0; off >>= 1) {
            if (tid < off) sRed[tid] += sRed[tid + off];
            __syncthreads();
        }
        result = sRed[0] * (1.0f / P_PTS);
    }

    if (tid == 0) per_anchor[m] = result;
}

__global__ void __launch_bounds__(M_ANCH)
mean_reduce_kernel(const float* __restrict__ per_anchor, float* __restrict__ out) {
    __shared__ float sRed[M_ANCH];
    const int tid = threadIdx.x;
    sRed[tid] = per_anchor[tid];
    __syncthreads();
    for (int off = M_ANCH / 2; off > 0; off >>= 1) {
        if (tid < off) sRed[tid] += sRed[tid + off];
        __syncthreads();
    }
    if (tid == 0) out[0] = sRed[0] * (1.0f / M_ANCH);
}

extern "C" void kernel_launch(void* const* d_in, const int* in_sizes, int n_in,
                              void* d_out, int out_size, void* d_ws, size_t ws_size,
                              hipStream_t stream) {
    const float* reg_rot      = (const float*)d_in[0];
    const float* reg_trans    = (const float*)d_in[1];
    const float* tgt_rot      = (const float*)d_in[2];
    const float* tgt_trans    = (const float*)d_in[3];
    const float* model_points = (const float*)d_in[4];
    const int*   is_sym       = (const int*)d_in[5];
    const int*   cls_idx      = (const int*)d_in[6];
    float* out = (float*)d_out;
    float* per_anchor = (float*)d_ws;  // 128 floats of scratch

    anchor_distance_kernel<<<M_ANCH, NTHREADS, 0, stream>>>(
        reg_rot, reg_trans, tgt_rot, tgt_trans, model_points, is_sym, cls_idx, per_anchor);
    mean_reduce_kernel<<<1, M_ANCH, 0, stream>>>(per_anchor, out);
}